// ReweightByTeacher_4329327035176
// MI455X (gfx1250) — compile-verified
//
#include <hip/hip_runtime.h>
#include <hip/hip_bf16.h>
#include <stdint.h>

// Problem constants (from reference)
constexpr int V      = 32000;
constexpr int NROWS  = 8192;

// Streaming geometry
constexpr int NW       = 10;            // wave32s per block
constexpr int NTHREADS = NW * 32;       // 320
constexpr int CHUNK_F  = 128;           // floats per chunk = 32 lanes * 4 (b128)
constexpr int NCHUNK   = V / (CHUNK_F * NW); // 25 chunks per wave, exact
constexpr int DEPTH    = 4;             // async ring depth per wave

static_assert(NCHUNK * CHUNK_F * NW == V, "geometry must tile V exactly");

// ---- CDNA5 async global->LDS helpers (gfx1250) ----------------------------

__device__ __forceinline__ void async_load_b128_nt(uint32_t lds_off, const void* gaddr) {
    // GV form: vdst = LDS byte address, v[addr:addr+1] = 64-bit global address.
    // NT hint: 1 GB read-once stream, keep it out of the way of the gathers in L2.
    asm volatile("global_load_async_to_lds_b128 %0, %1, off th:TH_LOAD_NT"
                 :: "v"(lds_off), "v"((uint64_t)(uintptr_t)gaddr)
                 : "memory");
}

__device__ __forceinline__ void async_refill_b128_nt(uint32_t lds_off, const void* gaddr) {
    // s_wait_dscnt 0 first: our ds_load of this slot must have landed in VGPRs
    // before the async engine may overwrite the slot.
    asm volatile("s_wait_dscnt 0\n\t"
                 "global_load_async_to_lds_b128 %0, %1, off th:TH_LOAD_NT"
                 :: "v"(lds_off), "v"((uint64_t)(uintptr_t)gaddr)
                 : "memory");
}

__device__ __forceinline__ void wait_async_le3() {
    asm volatile("s_wait_asynccnt 3" ::: "memory");
}
__device__ __forceinline__ void wait_async_le0() {
    asm volatile("s_wait_asynccnt 0" ::: "memory");
}

// ---- Kernel 1: one row per block; online softmax over async-streamed data --

__global__ __launch_bounds__(NTHREADS)
void ce_row_kernel(const float* __restrict__ logits,
                   const float* __restrict__ bias_probs,
                   const int*   __restrict__ labels,
                   float* __restrict__ wl_out,    // [NROWS] w_i * loss_i
                   float* __restrict__ w_out)     // [NROWS] w_i
{
    __shared__ __align__(16) float ring[NW * DEPTH * CHUNK_F]; // 20 KB
    __shared__ float rm[NW];
    __shared__ float rs[NW];

    const int row  = blockIdx.x;
    const int w    = threadIdx.x >> 5;   // wave id 0..9
    const int l    = threadIdx.x & 31;   // lane id
    const float* __restrict__ rowp = logits + (size_t)row * V;

    // Per-lane LDS float index of this wave's ring, slot 0.
    const int lds_lane = w * (DEPTH * CHUNK_F) + l * 4;

    // Prologue: fill the 4-deep ring.
#pragma unroll
    for (int k = 0; k < DEPTH; ++k) {
        uint32_t lo = (uint32_t)(uintptr_t)(&ring[lds_lane + k * CHUNK_F]);
        const float* g = rowp + (w + k * NW) * CHUNK_F + l * 4;
        async_load_b128_nt(lo, g);
    }

    float m = -3.402823466e38f; // running max
    float s = 0.0f;             // running sum of exp(x - m)

    // Steady state: 21 iterations, each consumes the oldest chunk and refills.
    for (int k = 0; k < NCHUNK - DEPTH; ++k) {
        wait_async_le3();                        // oldest chunk (k) has landed
        const int slot = k & (DEPTH - 1);
        const float4 x = *(const float4*)&ring[lds_lane + slot * CHUNK_F];

        const float mx = fmaxf(fmaxf(x.x, x.y), fmaxf(x.z, x.w));
        const float M  = fmaxf(m, mx);
        s = s * __expf(m - M)
          + __expf(x.x - M) + __expf(x.y - M) + __expf(x.z - M) + __expf(x.w - M);
        m = M;

        uint32_t lo = (uint32_t)(uintptr_t)(&ring[lds_lane + slot * CHUNK_F]);
        const float* g = rowp + (w + (k + DEPTH) * NW) * CHUNK_F + l * 4;
        async_refill_b128_nt(lo, g);
    }

    // Epilogue: drain the last DEPTH chunks.
    wait_async_le0();
#pragma unroll
    for (int k = NCHUNK - DEPTH; k < NCHUNK; ++k) {
        const int slot = k & (DEPTH - 1);
        const float4 x = *(const float4*)&ring[lds_lane + slot * CHUNK_F];
        const float mx = fmaxf(fmaxf(x.x, x.y), fmaxf(x.z, x.w));
        const float M  = fmaxf(m, mx);
        s = s * __expf(m - M)
          + __expf(x.x - M) + __expf(x.y - M) + __expf(x.z - M) + __expf(x.w - M);
        m = M;
    }

    // Wave32 merge of (m, s) pairs.
#pragma unroll
    for (int off = 16; off > 0; off >>= 1) {
        const float m2 = __shfl_xor(m, off, 32);
        const float s2 = __shfl_xor(s, off, 32);
        const float M  = fmaxf(m, m2);
        s = s * __expf(m - M) + s2 * __expf(m2 - M);
        m = M;
    }
    if (l == 0) { rm[w] = m; rs[w] = s; }
    __syncthreads();

    if (threadIdx.x == 0) {
        float M = rm[0], S = rs[0];
#pragma unroll
        for (int i = 1; i < NW; ++i) {
            const float M2 = fmaxf(M, rm[i]);
            S = S * __expf(M - M2) + rs[i] * __expf(rm[i] - M2);
            M = M2;
        }
        const int   lab  = labels[row];
        const float xl   = rowp[lab];
        const float wgt  = 1.0f - bias_probs[(size_t)row * V + lab];
        const float loss = (M + logf(S)) - xl;   // -log_softmax at the label
        wl_out[row] = wgt * loss;
        w_out[row]  = wgt;
    }
}

// ---- Kernel 2: deterministic final reduction of 8192 (w*loss, w) pairs ----

__global__ __launch_bounds__(256)
void ce_finalize_kernel(const float* __restrict__ wl,
                        const float* __restrict__ wsum,
                        float* __restrict__ out)
{
    __shared__ float sa[8];
    __shared__ float sb[8];
    float a = 0.0f, b = 0.0f;
    for (int i = threadIdx.x; i < NROWS; i += 256) {
        a += wl[i];
        b += wsum[i];
    }
#pragma unroll
    for (int off = 16; off > 0; off >>= 1) {
        a += __shfl_xor(a, off, 32);
        b += __shfl_xor(b, off, 32);
    }
    const int w = threadIdx.x >> 5, l = threadIdx.x & 31;
    if (l == 0) { sa[w] = a; sb[w] = b; }
    __syncthreads();
    if (threadIdx.x == 0) {
        float A = 0.0f, B = 0.0f;
#pragma unroll
        for (int i = 0; i < 8; ++i) { A += sa[i]; B += sb[i]; }
        out[0] = A / B;
    }
}

// ---- Host-side launcher ----------------------------------------------------

extern "C" void kernel_launch(void* const* d_in, const int* in_sizes, int n_in,
                              void* d_out, int out_size, void* d_ws, size_t ws_size,
                              hipStream_t stream) {
    const float* logits     = (const float*)d_in[0];
    const float* bias_probs = (const float*)d_in[1];
    const int*   labels     = (const int*)d_in[2];
    float*       out        = (float*)d_out;

    float* wl   = (float*)d_ws;        // [NROWS]
    float* wsum = wl + NROWS;          // [NROWS]

    ce_row_kernel<<<NROWS, NTHREADS, 0, stream>>>(logits, bias_probs, labels, wl, wsum);
    ce_finalize_kernel<<<1, 256, 0, stream>>>(wl, wsum, out);
}